// TopKGate_69561290326075
// MI455X (gfx1250) — compile-verified
//
#include <hip/hip_runtime.h>
#include <hip/hip_bf16.h>

typedef __attribute__((ext_vector_type(2))) float v2f;
typedef __attribute__((ext_vector_type(8))) float v8f;

#define DIM   4096
#define NEXP  64
#define STOK  16384

#define WAVES_PER_BLOCK  4
#define TOKENS_PER_WAVE  16
#define TOKENS_PER_BLOCK (WAVES_PER_BLOCK * TOKENS_PER_WAVE)   // 64
#define NBLOCKS          (STOK / TOKENS_PER_BLOCK)             // 256

// ---------------------------------------------------------------------------
// Zero the workspace accumulators (64 prob-sums + 64 argmax counts).
// ---------------------------------------------------------------------------
__global__ void moe_gate_zero_ws(float* __restrict__ ws) {
    const int i = threadIdx.x;
    if (i < 2 * NEXP) ws[i] = 0.0f;
}

// ---------------------------------------------------------------------------
// Main kernel: fp32 WMMA GEMM (16-token tile per wave) + softmax + top-2 +
// per-block aux-loss partial sums.
// ---------------------------------------------------------------------------
__global__ __launch_bounds__(128) void moe_gate_main(
    const float* __restrict__ x,     // [STOK, DIM]
    const float* __restrict__ W,     // [DIM, NEXP]
    const float* __restrict__ bias,  // [NEXP]
    float*       __restrict__ out,   // [4*STOK + 1]
    float*       __restrict__ ws)    // [2*NEXP] global accumulators
{
    // Per-wave 16x64 logits, stride 65 to avoid LDS bank conflicts on row scans.
    __shared__ float sLogits[WAVES_PER_BLOCK][16][65];
    __shared__ float sProb[NEXP];    // block-local sum of probs per expert
    __shared__ float sCount[NEXP];   // block-local argmax counts per expert

    const int tid  = threadIdx.x;
    const int wave = tid >> 5;
    const int lane = tid & 31;
    const int l16  = lane & 15;          // M (A) / N (B) index within tile
    const int kh   = (lane >> 4) * 2;    // half-wave K pair: {0,1} or {2,3}

    if (tid < NEXP) { sProb[tid] = 0.0f; sCount[tid] = 0.0f; }

    const int tokBase = blockIdx.x * TOKENS_PER_BLOCK + wave * TOKENS_PER_WAVE;
    const float* __restrict__ xrow = x + (size_t)(tokBase + l16) * DIM;

    v8f c0 = {}, c1 = {}, c2 = {}, c3 = {};

    for (int k0 = 0; k0 < DIM; k0 += 4) {
        if ((k0 & 127) == 0)  // prefetch x stream ahead (global_prefetch_b8)
            __builtin_prefetch(xrow + k0 + 256, 0, 3);

        const int k = k0 + kh;

        // A fragment: 16x4 f32, one b64 load per lane.
        v2f a;
        a.x = xrow[k];
        a.y = xrow[k + 1];

        // B fragments: 4x16 f32 per 16-expert tile.
        const float* __restrict__ wr0 = W + (size_t)k * NEXP;
        const float* __restrict__ wr1 = wr0 + NEXP;
        v2f b0, b1, b2, b3;
        b0.x = wr0[l16 +  0]; b0.y = wr1[l16 +  0];
        b1.x = wr0[l16 + 16]; b1.y = wr1[l16 + 16];
        b2.x = wr0[l16 + 32]; b2.y = wr1[l16 + 32];
        b3.x = wr0[l16 + 48]; b3.y = wr1[l16 + 48];

        c0 = __builtin_amdgcn_wmma_f32_16x16x4_f32(false, a, false, b0, (short)0, c0, false, false);
        c1 = __builtin_amdgcn_wmma_f32_16x16x4_f32(false, a, false, b1, (short)0, c1, false, false);
        c2 = __builtin_amdgcn_wmma_f32_16x16x4_f32(false, a, false, b2, (short)0, c2, false, false);
        c3 = __builtin_amdgcn_wmma_f32_16x16x4_f32(false, a, false, b3, (short)0, c3, false, false);
    }

    // Scatter accumulators (+bias) into LDS.
    // D layout: VGPR r -> row (halfwave*8 + r), col = tile*16 + l16.
    const int mbase = (lane >> 4) * 8;
#pragma unroll
    for (int r = 0; r < 8; ++r) {
        const int rowi = mbase + r;
        sLogits[wave][rowi][ 0 + l16] = c0[r] + bias[ 0 + l16];
        sLogits[wave][rowi][16 + l16] = c1[r] + bias[16 + l16];
        sLogits[wave][rowi][32 + l16] = c2[r] + bias[32 + l16];
        sLogits[wave][rowi][48 + l16] = c3[r] + bias[48 + l16];
    }
    __syncthreads();

    // Softmax + top-2 per token: lanes 0..15 each own one token (64 experts).
    if (lane < 16) {
        const int tok = tokBase + lane;
        float* lg = &sLogits[wave][lane][0];

        float mx = lg[0];
#pragma unroll
        for (int e = 1; e < NEXP; ++e) mx = fmaxf(mx, lg[e]);

        float sum = 0.0f;
#pragma unroll
        for (int e = 0; e < NEXP; ++e) {
            const float t = __expf(lg[e] - mx);
            lg[e] = t;
            sum += t;
        }
        const float inv = 1.0f / sum;

        float v1 = -1.0f; int i1 = 0;
#pragma unroll
        for (int e = 0; e < NEXP; ++e) {
            const float p = lg[e] * inv;
            lg[e] = p;
            if (p > v1) { v1 = p; i1 = e; }      // first-index tie break
            atomicAdd(&sProb[e], p);             // importance partial
        }
        float v2 = -1.0f; int i2 = 0;
#pragma unroll
        for (int e = 0; e < NEXP; ++e) {
            if (e == i1) continue;
            const float p = lg[e];
            if (p > v2) { v2 = p; i2 = e; }
        }
        atomicAdd(&sCount[i1], 1.0f);            // hard-routing load partial

        // Outputs: idx region [0, 2S) as floats, vals region [2S, 4S).
        out[2 * tok + 0]            = (float)i1;
        out[2 * tok + 1]            = (float)i2;
        out[2 * STOK + 2 * tok + 0] = v1;
        out[2 * STOK + 2 * tok + 1] = v2;
    }
    __syncthreads();

    // Flush block partials to global workspace.
    if (tid < NEXP) {
        atomicAdd(&ws[tid],        sProb[tid]);
        atomicAdd(&ws[NEXP + tid], sCount[tid]);
    }
}

// ---------------------------------------------------------------------------
// Finalize: aux_loss = E * sum_e( importance_e * max(load_e, 1e-9) )
// ---------------------------------------------------------------------------
__global__ void moe_gate_finalize(const float* __restrict__ ws,
                                  float* __restrict__ out) {
    __shared__ float red[NEXP];
    const int e = threadIdx.x;            // 64 threads
    const float imp  = ws[e]        / (float)STOK;
    const float load = ws[NEXP + e] / (float)STOK;
    red[e] = imp * fmaxf(load, 1e-9f);
    __syncthreads();
#pragma unroll
    for (int s = NEXP / 2; s > 0; s >>= 1) {
        if (e < s) red[e] += red[e + s];
        __syncthreads();
    }
    if (e == 0) out[4 * STOK] = (float)NEXP * red[0];
}

// ---------------------------------------------------------------------------
extern "C" void kernel_launch(void* const* d_in, const int* in_sizes, int n_in,
                              void* d_out, int out_size, void* d_ws, size_t ws_size,
                              hipStream_t stream) {
    const float* x    = (const float*)d_in[0];   // [STOK, DIM]
    const float* W    = (const float*)d_in[1];   // [DIM, NEXP]
    const float* bias = (const float*)d_in[2];   // [NEXP]
    float* out = (float*)d_out;
    float* ws  = (float*)d_ws;

    moe_gate_zero_ws<<<1, 128, 0, stream>>>(ws);
    moe_gate_main<<<NBLOCKS, 128, 0, stream>>>(x, W, bias, out, ws);
    moe_gate_finalize<<<1, NEXP, 0, stream>>>(ws, out);
}